// FreedriftModel_37744172597621
// MI455X (gfx1250) — compile-verified
//
#include <hip/hip_runtime.h>
#include <hip/hip_bf16.h>
#include <math.h>

// ---------------------------------------------------------------------------
// Freedrift semi-Lagrangian model, restructured for MI455X (gfx1250):
//  - conv precompute as fp32 WMMA (V_WMMA_F32_16X16X4_F32) over 18 frames
//  - 576-step walk parallelized across the 4096 independent tracers
//  - all intermediates (~640KB) L2-resident
// ---------------------------------------------------------------------------

typedef __attribute__((ext_vector_type(2))) float v2f;
typedef __attribute__((ext_vector_type(8))) float v8f;

#define GRID 64
#define NPIX 4096           // 64*64
#define NFRAMES 17
#define NSTEPS 576          // (17-1)*36
// workspace layout (float offsets)
#define WS_OCEAN 0                       // 2*4096 : mean ocean channels
#define WS_CA    (2*NPIX)                // 18*2*4096 : conv fields (frame 17 = ocean conv + bias)
#define WS_DIST  (WS_CA + 18*2*NPIX)     // 2*4096 : distances
// total = 163840 floats = 640 KB

// --- Kernel 1: ocean feature mean + mesh distance fields --------------------
__global__ void fd_prep(const float* __restrict__ forcings,
                        const float* __restrict__ mesh,
                        float* __restrict__ ws) {
  int tid = blockIdx.x * blockDim.x + threadIdx.x;
  if (tid >= 2 * NPIX) return;
  int c = tid / NPIX;
  int g = tid - c * NPIX;
  int j = g >> 6, i = g & 63;

  // ocean_feat = mean over 17 frames of forcings[:, 2+c]
  float s = 0.0f;
  for (int t = 0; t < NFRAMES; ++t)
    s += forcings[(t * 4 + 2 + c) * NPIX + g];
  ws[WS_OCEAN + c * NPIX + g] = s * (1.0f / 17.0f);

  // distances, replicating _estimate_distances pad/concat indexing for the
  // [0:64,0:64] window that is ever gathered.
  const float* m0 = mesh;
  const float* m1 = mesh + NPIX;
  float d;
  if (c == 0) { // dx field: dxs[max(j-1,0)][clamp(i-1,0,62)]
    int jj = (j > 0) ? (j - 1) : 0;
    int ii = i - 1; if (ii < 0) ii = 0; if (ii > 62) ii = 62;
    float e0 = m0[jj * GRID + ii + 1] - m0[jj * GRID + ii];
    float e1 = m1[jj * GRID + ii + 1] - m1[jj * GRID + ii];
    d = sqrtf(e0 * e0 + e1 * e1);
  } else {      // dy field: dys[clamp(j-1,0,62)][max(i-1,0)]
    int ii = (i > 0) ? (i - 1) : 0;
    int jj = j - 1; if (jj < 0) jj = 0; if (jj > 62) jj = 62;
    float e0 = m0[(jj + 1) * GRID + ii] - m0[jj * GRID + ii];
    float e1 = m1[(jj + 1) * GRID + ii] - m1[jj * GRID + ii];
    d = sqrtf(e0 * e0 + e1 * e1);
  }
  ws[WS_DIST + c * NPIX + g] = d;
}

// --- Kernel 2: 3x3 conv (2 in-ch -> 2 out-ch, K=18 taps) as fp32 WMMA -------
// Frames 0..16: atm channels of forcings[f], weight in-channels 0..1.
// Frame  17   : ocean_feat, weight in-channels 2..3, plus bias.
// Per wave: one 16-pixel tile. M=oc (rows 0,1 used), N=16 pixels, K=18 pad 20.
__global__ void fd_conv_wmma(const float* __restrict__ forcings,
                             const float* __restrict__ conv_w,
                             const float* __restrict__ conv_b,
                             float* __restrict__ ws) {
  int f = blockIdx.y;                  // 0..17
  int wave = threadIdx.x >> 5;
  int lane = threadIdx.x & 31;
  int tile = blockIdx.x * 8 + wave;    // 0..255
  int pbase = tile * 16;
  int nm = lane & 15;                  // M index for A, N index for B/D
  int hi = lane >> 4;

  const float* in0;
  const float* in1;
  int wcb;
  if (f < NFRAMES) {
    in0 = forcings + (f * 4 + 0) * NPIX;
    in1 = forcings + (f * 4 + 1) * NPIX;
    wcb = 0;
  } else {
    in0 = ws + WS_OCEAN;
    in1 = ws + WS_OCEAN + NPIX;
    wcb = 2;
  }

  int p = pbase + nm;
  int py = p >> 6, px = p & 63;

  v8f acc = {0.f, 0.f, 0.f, 0.f, 0.f, 0.f, 0.f, 0.f};
#pragma unroll
  for (int kc = 0; kc < 5; ++kc) {
    v2f a, b;
#pragma unroll
    for (int v = 0; v < 2; ++v) {
      // ISA 16x4 f32 A layout: lanes 0-15 K=0(V0),1(V1); lanes 16-31 K=2,3.
      int tap = kc * 4 + hi * 2 + v;
      float av = 0.0f, bv = 0.0f;
      if (tap < 18) {
        int ic = tap / 9;
        int r  = tap - ic * 9;
        int ky = r / 3, kx = r - ky * 3;
        if (nm < 2)
          av = conv_w[((nm * 4 + wcb + ic) * 3 + ky) * 3 + kx];
        int iy = py + ky - 1, ix = px + kx - 1;
        if (iy >= 0 && iy < GRID && ix >= 0 && ix < GRID)
          bv = (ic == 0 ? in0 : in1)[iy * GRID + ix];
      }
      a[v] = av;
      b[v] = bv;
    }
    acc = __builtin_amdgcn_wmma_f32_16x16x4_f32(
        /*neg_a=*/false, a, /*neg_b=*/false, b,
        /*c_mod=*/(short)0, acc, /*reuse_a=*/false, /*reuse_b=*/false);
  }

  // D layout: lanes 0-15, VGPR0 = row M=0, VGPR1 = row M=1 (our 2 out-chans)
  if (hi == 0) {
    float o0 = acc[0], o1 = acc[1];
    if (f == NFRAMES) { o0 += conv_b[0]; o1 += conv_b[1]; }
    ws[WS_CA + (f * 2 + 0) * NPIX + p] = o0;
    ws[WS_CA + (f * 2 + 1) * NPIX + p] = o1;
  }
}

// --- Kernel 3: per-tracer 576-step walk + final bilinear interp -------------
__global__ void fd_tracer(const float* __restrict__ init_cond,
                          const float* __restrict__ ws,
                          float* __restrict__ out) {
  int i = blockIdx.x * blockDim.x + threadIdx.x;
  if (i >= NPIX) return;

  float px = (float)(i & 63);
  float py = (float)(i >> 6);
  float vx = px, vy = py;

  const float* CA = ws + WS_CA;                      // [18][2][4096]
  const float* CO = ws + WS_CA + NFRAMES * 2 * NPIX; // frame 17
  const float* D0 = ws + WS_DIST;
  const float* D1 = ws + WS_DIST + NPIX;
  const float inv_norm = 1.0f / (1.0f + 2e-9f);

  for (int step = NSTEPS - 1; step >= 0; --step) {
    float t  = ((float)step + 0.5f) / 36.0f;   // never an integer
    float tf = floorf(t);
    int   fi = (int)tf;
    float w1 = (t - tf + 1e-9f) * inv_norm;
    float w0 = ((tf + 1.0f) - t + 1e-9f) * inv_norm;

    float cx = fminf(fmaxf(vx, 0.0f), 63.0f);
    float cy = fminf(fmaxf(vy, 0.0f), 63.0f);
    int gx = (int)rintf(cx);                   // round-half-even == jnp.round
    int gy = (int)rintf(cy);
    int g = gy * GRID + gx;

    int bf = fi * 2 * NPIX;
    float y0 = w0 * CA[bf + g]            + w1 * CA[bf + 2 * NPIX + g]        + CO[g];
    float y1 = w0 * CA[bf + NPIX + g]     + w1 * CA[bf + 3 * NPIX + g]        + CO[NPIX + g];

    float v0 = 0.2f * tanhf(y0) / D0[g];
    float v1 = 0.2f * tanhf(y1) / D1[g];
    float dx = -600.0f * v0;                   // DT_MODEL = 600
    float dy = -600.0f * v1;
    px += dx; py += dy;
    vx = px + 1.5f * dx;
    vy = py + 1.5f * dy;
  }

  // bilinear interpolation of the 8 channels at final position
  float tx = fminf(fmaxf(px, 0.0f), 63.0f);
  float ty = fminf(fmaxf(py, 0.0f), 63.0f);
  float flx = floorf(tx), fly = floorf(ty);
  int fx = (int)flx, fy = (int)fly;
  int cx2 = (int)ceilf(tx), cy2 = (int)ceilf(ty);
  float nx = tx - flx, ny = ty - fly;
#pragma unroll
  for (int ch = 0; ch < 8; ++ch) {
    const float* s = init_cond + ch * NPIX;
    float a00 = s[fy * GRID + fx];
    float a10 = s[fy * GRID + cx2] - a00;
    float a01 = s[cy2 * GRID + fx] - a00;
    float a11 = s[cy2 * GRID + cx2] - s[fy * GRID + cx2] - s[cy2 * GRID + fx] + a00;
    out[ch * NPIX + i] = a00 + a10 * nx + a01 * ny + a11 * nx * ny;
  }
}

extern "C" void kernel_launch(void* const* d_in, const int* in_sizes, int n_in,
                              void* d_out, int out_size, void* d_ws, size_t ws_size,
                              hipStream_t stream) {
  const float* init_cond = (const float*)d_in[0]; // (8,64,64)
  const float* forcings  = (const float*)d_in[1]; // (17,4,64,64)
  const float* mesh      = (const float*)d_in[2]; // (2,64,64)
  const float* conv_w    = (const float*)d_in[3]; // (2,4,3,3)
  const float* conv_b    = (const float*)d_in[4]; // (2,)
  float* ws = (float*)d_ws;                       // needs 640 KB

  fd_prep<<<dim3(32), dim3(256), 0, stream>>>(forcings, mesh, ws);
  fd_conv_wmma<<<dim3(32, 18), dim3(256), 0, stream>>>(forcings, conv_w, conv_b, ws);
  fd_tracer<<<dim3(16), dim3(256), 0, stream>>>(init_cond, ws, (float*)d_out);
}